// RandScatterRouter_80427557585600
// MI455X (gfx1250) — compile-verified
//
#include <hip/hip_runtime.h>
#include <stdint.h>

// ---------------------------------------------------------------------------
// Top-1 scatter router for MI455X (gfx1250).
//   path   = argmax(score, -1)           [N]
//   order  = stable argsort(path)        [N]
//   out    = inputs[order]               [N, D]
//   counts = histogram(path, 8)          [8]
// Memory-bound: ~1.07 GB traffic -> ~46 us at 23.3 TB/s. No matmul, so no
// WMMA; the CDNA5-specific path exercised is the async global<->LDS data
// mover (global_load_async_to_lds_b128 / global_store_async_from_lds_b128,
// tracked by ASYNCcnt) for the 4 KB row payloads.
// ---------------------------------------------------------------------------

#define NT  131072      // tokens
#define DM  1024        // d_model
#define NP  8           // paths
#define BLK 256         // tokens per histogram block
#define NB  (NT / BLK)  // 512 blocks

// Async global<->LDS builtins exist only on the gfx1250 device pass; the host
// pass (and any toolchain lacking them) uses the float4 register-copy path.
#if defined(__AMDGCN__) &&                                                   \
    __has_builtin(__builtin_amdgcn_global_load_async_to_lds_b128) &&         \
    __has_builtin(__builtin_amdgcn_global_store_async_from_lds_b128)
#define USE_ASYNC_LDS 1
#else
#define USE_ASYNC_LDS 0
#endif

typedef __attribute__((address_space(1))) char g_char;  // global byte ptr
typedef __attribute__((address_space(3))) char l_char;  // LDS byte ptr

typedef int v4i __attribute__((ext_vector_type(4)));
typedef __attribute__((address_space(1))) v4i* g4p;  // global int4* (AS1)
typedef __attribute__((address_space(3))) v4i* l4p;  // LDS int4* (AS3)

__device__ __forceinline__ void wait_async0() {
#if defined(__AMDGCN__)
#if __has_builtin(__builtin_amdgcn_s_wait_asynccnt)
  __builtin_amdgcn_s_wait_asynccnt(0);
#else
  asm volatile("s_wait_asynccnt 0" ::: "memory");
#endif
#endif
}

// ---------------------------------------------------------------------------
// Kernel 1: per-token argmax + stable within-block rank + block histograms.
// One thread per token, 256 tokens per block (8 wave32 waves).
// ---------------------------------------------------------------------------
__global__ void __launch_bounds__(BLK) k_path(const float* __restrict__ score,
                                              uint32_t* __restrict__ pr,
                                              uint32_t* __restrict__ blockCounts) {
  const uint32_t t = blockIdx.x * BLK + threadIdx.x;

  // 8 scores per token, 32B aligned -> two float4 loads.
  const float4* s4 = (const float4*)(score + (size_t)t * NP);
  const float4 a = s4[0];
  const float4 b = s4[1];
  const float v[NP] = {a.x, a.y, a.z, a.w, b.x, b.y, b.z, b.w};
  int p = 0;
  float best = v[0];
#pragma unroll
  for (int q = 1; q < NP; ++q) {
    if (v[q] > best) { best = v[q]; p = q; }  // strict '>' == jnp first-max
  }

  const uint32_t lane = threadIdx.x & 31u;
  const uint32_t wave = threadIdx.x >> 5;
  __shared__ uint32_t waveHist[BLK / 32][NP];

  // wave32 ballots: one mask per path.
  unsigned long long ball[NP];
#pragma unroll
  for (int q = 0; q < NP; ++q) ball[q] = __ballot(p == q);

  const uint32_t rank_in_wave =
      (uint32_t)__popcll(ball[p] & ((1ull << lane) - 1ull));
  if (lane < NP) waveHist[wave][lane] = (uint32_t)__popcll(ball[lane]);
  __syncthreads();

  uint32_t prefix = 0;
  for (uint32_t w = 0; w < wave; ++w) prefix += waveHist[w][p];
  const uint32_t rank = prefix + rank_in_wave;  // < 256, fits in 8 bits

  pr[t] = ((uint32_t)p << 8) | rank;

  if (threadIdx.x < NP) {
    uint32_t tot = 0;
#pragma unroll
    for (int w = 0; w < BLK / 32; ++w) tot += waveHist[w][threadIdx.x];
    blockCounts[blockIdx.x * NP + threadIdx.x] = tot;
  }
}

// ---------------------------------------------------------------------------
// Kernel 2: tiny scan. totals per path -> exclusive path bases -> per-(block,
// path) destination bases. Single block; 8 working lanes, 512 iters each.
// ---------------------------------------------------------------------------
__global__ void k_scan(const uint32_t* __restrict__ blockCounts,
                       uint32_t* __restrict__ destBase,
                       float* __restrict__ countsf) {
  __shared__ uint32_t totals[NP];
  __shared__ uint32_t base[NP];
  const uint32_t p = threadIdx.x;

  if (p < NP) {
    uint32_t s = 0;
    for (int b = 0; b < NB; ++b) s += blockCounts[b * NP + p];
    totals[p] = s;
  }
  __syncthreads();
  if (threadIdx.x == 0) {
    uint32_t acc = 0;
    for (int q = 0; q < NP; ++q) { base[q] = acc; acc += totals[q]; }
  }
  __syncthreads();
  if (p < NP) {
    uint32_t run = base[p];
    for (int b = 0; b < NB; ++b) {
      destBase[b * NP + p] = run;
      run += blockCounts[b * NP + p];
    }
    countsf[p] = (float)totals[p];
  }
}

// ---------------------------------------------------------------------------
// Kernel 3: dispatch. One wave per token row (8 rows per 256-thread block).
// Row copy (4 KB) goes through the CDNA5 async data mover: 8x async B128
// load-to-LDS (512 B/wave each), s_wait_asynccnt 0, 8x async B128
// store-from-LDS. Implicit wait-idle at s_endpgm covers the stores.
// ---------------------------------------------------------------------------
__global__ void __launch_bounds__(256) k_dispatch(
    const float* __restrict__ in, const uint32_t* __restrict__ pr,
    const uint32_t* __restrict__ destBase, float* __restrict__ out,
    float* __restrict__ orderf) {
#if USE_ASYNC_LDS
  __shared__ float lds[8 * DM];  // 4 KB per wave, 32 KB per block
#endif
  const uint32_t wave = threadIdx.x >> 5;
  const uint32_t lane = threadIdx.x & 31u;
  const uint32_t t = blockIdx.x * 8 + wave;

  const uint32_t e = pr[t];
  const uint32_t p = e >> 8;
  const uint32_t rank = e & 0xFFu;
  const uint32_t dest = destBase[(t >> 8) * NP + p] + rank;

  if (lane == 0) orderf[dest] = (float)t;

  const float* src = in + (size_t)t * DM;
  float* dst = out + (size_t)dest * DM;

#if USE_ASYNC_LDS
  l_char* lbase = (l_char*)(&lds[wave * DM]) + lane * 16;
  g_char* gsrc = (g_char*)src + lane * 16;
  g_char* gdst = (g_char*)dst + lane * 16;
#pragma unroll
  for (int k = 0; k < 8; ++k)
    __builtin_amdgcn_global_load_async_to_lds_b128(
        (g4p)(gsrc + k * 512), (l4p)(lbase + k * 512), 0, 0);
  wait_async0();  // loads must land in LDS before stores read it
#pragma unroll
  for (int k = 0; k < 8; ++k)
    __builtin_amdgcn_global_store_async_from_lds_b128(
        (g4p)(gdst + k * 512), (l4p)(lbase + k * 512), 0, 0);
#else
  // Fallback: straight register copy, 8 outstanding B128 loads per lane.
  const float4* s4 = (const float4*)src;
  float4* d4 = (float4*)dst;
  float4 r[8];
#pragma unroll
  for (int k = 0; k < 8; ++k) r[k] = s4[lane + 32 * k];
#pragma unroll
  for (int k = 0; k < 8; ++k) d4[lane + 32 * k] = r[k];
#endif
}

// ---------------------------------------------------------------------------
// Host entry. d_in[0]=inputs [N*D] f32, d_in[1]=score [N*8] f32.
// d_out = dispatched (N*D) | order (N, as f32) | counts (8, as f32).
// d_ws needs (NT + 2*NB*NP)*4 = 544 KB.
// ---------------------------------------------------------------------------
extern "C" void kernel_launch(void* const* d_in, const int* in_sizes, int n_in,
                              void* d_out, int out_size, void* d_ws,
                              size_t ws_size, hipStream_t stream) {
  (void)in_sizes; (void)n_in; (void)out_size; (void)ws_size;

  const float* inputs = (const float*)d_in[0];
  const float* score = (const float*)d_in[1];

  uint32_t* ws = (uint32_t*)d_ws;
  uint32_t* pr = ws;                           // NT u32
  uint32_t* blockCounts = pr + NT;             // NB*NP u32
  uint32_t* destBase = blockCounts + NB * NP;  // NB*NP u32

  float* out = (float*)d_out;
  float* orderf = out + (size_t)NT * DM;
  float* countsf = orderf + NT;

  k_path<<<NB, BLK, 0, stream>>>(score, pr, blockCounts);
  k_scan<<<1, 32, 0, stream>>>(blockCounts, destBase, countsf);
  k_dispatch<<<NT / 8, 256, 0, stream>>>(inputs, pr, destBase, out, orderf);
}